// UniformAffineQuantizer_1812476199613
// MI455X (gfx1250) — compile-verified
//
#include <hip/hip_runtime.h>
#include <stdint.h>

// ---------------------------------------------------------------------------
// 4-bit group fake-quantizer, group=128 along last dim, fp32 in/out.
// One wave32 == one group (32 lanes x float4). TDM-staged, double-buffered.
// Cross-lane min/max via DS_SWIZZLE immediate xor patterns (no index VALU).
// ---------------------------------------------------------------------------

#define GROUP            128
#define CHUNK_FLOATS     4096                       // 32 groups = 16 KB per chunk
#define GROUPS_PER_CHUNK (CHUNK_FLOATS / GROUP)     // 32
#define THREADS          256                        // 8 waves (wave32)
#define WAVES            (THREADS / 32)
#define GROUPS_PER_WAVE  (GROUPS_PER_CHUNK / WAVES) // 4

typedef float        f32x4 __attribute__((ext_vector_type(4)));
typedef unsigned int v4u   __attribute__((ext_vector_type(4)));
typedef int          v4i   __attribute__((ext_vector_type(4)));
typedef int          v8i   __attribute__((ext_vector_type(8)));

#if defined(__HIP_DEVICE_COMPILE__) && __has_builtin(__builtin_amdgcn_tensor_load_to_lds)
#define USE_TDM 1
#else
#define USE_TDM 0
#endif

// ---- one butterfly step: ds_swizzle group-of-32, xor_mask=MASK, and_mask=0x1f
template <int MASK>
__device__ __forceinline__ void swz_minmax(float& mn, float& mx) {
    constexpr int pat = (MASK << 5) | 0x1f;   // {or=0, xor=MASK, and=0x1f}
    float omn = __int_as_float(__builtin_amdgcn_ds_swizzle(__float_as_int(mn), pat));
    float omx = __int_as_float(__builtin_amdgcn_ds_swizzle(__float_as_int(mx), pat));
    mn = fminf(mn, omn);
    mx = fmaxf(mx, omx);
}

// ---- per-group quantize: lane holds 4 contiguous elements of a 128-group ---
__device__ __forceinline__ f32x4 fakequant_group(f32x4 v) {
    float lmin = fminf(fminf(v.x, v.y), fminf(v.z, v.w));
    float lmax = fmaxf(fmaxf(v.x, v.y), fmaxf(v.z, v.w));
    swz_minmax<16>(lmin, lmax);
    swz_minmax<8>(lmin, lmax);
    swz_minmax<4>(lmin, lmax);
    swz_minmax<2>(lmin, lmax);
    swz_minmax<1>(lmin, lmax);
    float scale = (lmax - lmin) * (1.0f / 15.0f);           // /QMAX
    scale = fminf(fmaxf(scale, 1e-5f), 1e4f);               // clip scale
    float zp  = fminf(fmaxf(-lmin / scale, -1e4f), 1e4f);   // clip zp
    float rzp = rintf(zp);                                  // round-half-even
    float inv = 1.0f / scale;
    f32x4 o;
    o.x = (fminf(fmaxf(rintf(v.x * inv) + rzp, 0.0f), 15.0f) - rzp) * scale;
    o.y = (fminf(fmaxf(rintf(v.y * inv) + rzp, 0.0f), 15.0f) - rzp) * scale;
    o.z = (fminf(fmaxf(rintf(v.z * inv) + rzp, 0.0f), 15.0f) - rzp) * scale;
    o.w = (fminf(fmaxf(rintf(v.w * inv) + rzp, 0.0f), 15.0f) - rzp) * scale;
    return o;
}

#if USE_TDM
// generic -> LDS(addrspace 3) pointer, whose integer value is the LDS byte offset
__device__ __forceinline__ uint32_t lds_byte_offset(void* p) {
    return (uint32_t)(uintptr_t)(__attribute__((address_space(3))) char*)p;
}

// Issue one TDM descriptor: 1-D tile of CHUNK_FLOATS fp32 from global -> LDS.
// D# packing per CDNA5 ISA.
__device__ __forceinline__ void tdm_load_chunk(const float* gsrc, uint32_t lds_addr) {
    uint64_t ga = (uint64_t)(uintptr_t)gsrc;
    v4u g0;
    g0[0] = 1u;                                   // count=1, is_restore=0, gather=0
    g0[1] = lds_addr;                             // lds_addr (bytes)
    g0[2] = (uint32_t)ga;                         // global_addr[31:0]
    g0[3] = (uint32_t)(ga >> 32) | (2u << 30);    // global_addr[56:32] | type=2
    v8i g1;
    g1[0] = (int)(2u << 16);                               // data_size=2 -> 4 bytes
    g1[1] = (int)((CHUNK_FLOATS & 0xFFFF) << 16);          // tensor_dim0[15:0]
    g1[2] = (int)(((CHUNK_FLOATS >> 16) & 0xFFFF)          // tensor_dim0[31:16]
                 | (1u << 16));                            // tensor_dim1 = 1
    g1[3] = (int)((CHUNK_FLOATS & 0xFFFF) << 16);          // tile_dim0 = 4096
    g1[4] = 0;                                             // tile_dim1=0, tile_dim2=0
    g1[5] = CHUNK_FLOATS;                                  // tensor_dim0_stride lo32
    g1[6] = (int)((CHUNK_FLOATS & 0xFFFF) << 16);          // stride0 hi16 | stride1 lo16
    g1[7] = 0;                                             // stride1 hi32
    v4i z4 = {0, 0, 0, 0};
#if defined(__clang_major__) && (__clang_major__ >= 23)
    v8i z8 = {0, 0, 0, 0, 0, 0, 0, 0};
    __builtin_amdgcn_tensor_load_to_lds(g0, g1, z4, z4, z8, 0);
#else
    __builtin_amdgcn_tensor_load_to_lds(g0, g1, z4, z4, 0);
#endif
}
#endif // USE_TDM

__global__ __launch_bounds__(THREADS)
void fakequant4_g128_kernel(const float* __restrict__ x,
                            float* __restrict__ out,
                            int nChunks) {
    const int lane = threadIdx.x & 31;
    const int wave = threadIdx.x >> 5;

#if USE_TDM
    __shared__ __align__(16) float smem[2][CHUNK_FLOATS];
    int c = (int)blockIdx.x;
    if (c >= nChunks) return;                      // block-uniform early out
    int buf = 0;
    if (wave == 0)
        tdm_load_chunk(x + (size_t)c * CHUNK_FLOATS, lds_byte_offset(&smem[0][0]));
    for (;;) {
        const int  cn      = c + (int)gridDim.x;
        const bool hasNext = cn < nChunks;
        if (wave == 0) {
            if (hasNext) {
                // prefetch next chunk into the other buffer, then make sure the
                // current chunk (older of the two in-order TDM ops) has landed
                tdm_load_chunk(x + (size_t)cn * CHUNK_FLOATS,
                               lds_byte_offset(&smem[buf ^ 1][0]));
                __builtin_amdgcn_s_wait_tensorcnt(1);
            } else {
                __builtin_amdgcn_s_wait_tensorcnt(0);
            }
        }
        __syncthreads();                           // chunk c visible to all waves
        #pragma unroll
        for (int j = 0; j < GROUPS_PER_WAVE; ++j) {
            const int g = wave * GROUPS_PER_WAVE + j;
            f32x4 v = *(const f32x4*)&smem[buf][g * GROUP + lane * 4];
            f32x4 o = fakequant_group(v);
            __builtin_nontemporal_store(
                o, (f32x4*)(out + (size_t)c * CHUNK_FLOATS + g * GROUP + lane * 4));
        }
        __syncthreads();                           // buffer reusable before next issue
        if (!hasNext) break;
        c = cn;
        buf ^= 1;
    }
#else
    // Fallback: direct streaming loads (still 128-bit, nontemporal, coalesced)
    const size_t nGroups = (size_t)nChunks * GROUPS_PER_CHUNK;
    for (size_t grp = (size_t)blockIdx.x * WAVES + wave; grp < nGroups;
         grp += (size_t)gridDim.x * WAVES) {
        const size_t base = grp * GROUP + (size_t)lane * 4;
        f32x4 v = __builtin_nontemporal_load((const f32x4*)(x + base));
        f32x4 o = fakequant_group(v);
        __builtin_nontemporal_store(o, (f32x4*)(out + base));
    }
#endif
}

extern "C" void kernel_launch(void* const* d_in, const int* in_sizes, int n_in,
                              void* d_out, int out_size, void* d_ws, size_t ws_size,
                              hipStream_t stream) {
    (void)n_in; (void)out_size; (void)d_ws; (void)ws_size;
    const float* x   = (const float*)d_in[0];
    float*       out = (float*)d_out;
    const long long n = (long long)in_sizes[0];            // 8192*8192
    const int nChunks = (int)(n / CHUNK_FLOATS);           // 16384
    int blocks = nChunks < 4096 ? nChunks : 4096;          // 4 chunks/block pipeline
    if (blocks < 1) blocks = 1;
    fakequant4_g128_kernel<<<dim3(blocks), dim3(THREADS), 0, stream>>>(x, out, nChunks);
}